// ETP_55207509623278
// MI455X (gfx1250) — compile-verified
//
#include <hip/hip_runtime.h>
#include <hip/hip_bf16.h>
#include <stdint.h>

typedef __attribute__((ext_vector_type(2))) float v2f;
typedef __attribute__((ext_vector_type(4))) float v4f;
typedef __attribute__((ext_vector_type(8))) float v8f;

#define N_DIM 4096
#define K_DIM 1024
#define TILE_M 128
#define TILE_N 128
#define TILE_K 16
#define LDSS 20   // padded stride: float4-aligned, 20*r mod 64 distinct for r=0..15

// ---------------------------------------------------------------------------
// sim = (X @ Y^T) / 32 via V_WMMA_F32_16X16X4_F32, double-buffered LDS tiles
// filled with GLOBAL_LOAD_ASYNC_TO_LDS_B128 (ASYNCcnt-tracked, VGPR-bypassing)
// ---------------------------------------------------------------------------
__global__ __launch_bounds__(256)
void ETP_sim_gemm(const float* __restrict__ X, const float* __restrict__ Y,
                  float* __restrict__ S) {
    __shared__ float As[2][TILE_M][LDSS];   // 2 x 10 KB
    __shared__ float Bs[2][TILE_N][LDSS];   // 2 x 10 KB

    const int tid  = threadIdx.x;
    const int wave = tid >> 5;
    const int lane = tid & 31;
    const int bm = blockIdx.x * TILE_M;
    const int bn = blockIdx.y * TILE_N;

    // 2x4 wave grid: each wave owns a 64x32 sub-tile = 4x2 fragments
    const int wm0 = (wave & 1) * 64;
    const int wn0 = (wave >> 1) * 32;
    const int lrow = lane & 15;          // M (or N) index within fragment
    const int lk2  = (lane >> 4) * 2;    // K base: lanes 0-15 -> {0,1}, 16-31 -> {2,3}

    // staging decomposition: 16B per lane, 128 rows x 4 segs = 512 slots, 2/thread
    const int r0  = tid >> 2;            // 0..63
    const int seg = (tid & 3) * 4;       // 0,4,8,12

    v8f acc[4][2] = {};

    auto stage = [&](int buf, int k0) {
        #pragma unroll
        for (int it = 0; it < 2; ++it) {
            int r = r0 + it * 64;
            uint32_t la = (uint32_t)(uintptr_t)&As[buf][r][seg];
            const float* ga = X + (size_t)(bm + r) * K_DIM + k0 + seg;
            asm volatile("global_load_async_to_lds_b128 %0, %1, off"
                         :: "v"(la), "v"(ga) : "memory");
            uint32_t lb = (uint32_t)(uintptr_t)&Bs[buf][r][seg];
            const float* gb = Y + (size_t)(bn + r) * K_DIM + k0 + seg;
            asm volatile("global_load_async_to_lds_b128 %0, %1, off"
                         :: "v"(lb), "v"(gb) : "memory");
        }
    };

    stage(0, 0);                               // prologue: fill buffer 0

    const int NT = K_DIM / TILE_K;             // 64 k-tiles
    for (int t = 0; t < NT; ++t) {
        const int cur = t & 1;

        if (t + 1 < NT) {
            stage(cur ^ 1, (t + 1) * TILE_K);  // async-fill next buffer
            // 4 new ops outstanding; waiting <=4 retires the 4 for 'cur'
            asm volatile("s_wait_asynccnt 4" ::: "memory");
        } else {
            asm volatile("s_wait_asynccnt 0" ::: "memory");
        }
        __syncthreads();                       // publish all waves' LDS writes

        #pragma unroll
        for (int kk = 0; kk < TILE_K; kk += 4) {
            v2f a[4], b[2];
            #pragma unroll
            for (int fm = 0; fm < 4; ++fm) {
                const float* p = &As[cur][wm0 + fm * 16 + lrow][kk + lk2];
                v2f tv = {p[0], p[1]};
                a[fm] = tv;
            }
            #pragma unroll
            for (int fn = 0; fn < 2; ++fn) {
                const float* p = &Bs[cur][wn0 + fn * 16 + lrow][kk + lk2];
                v2f tv = {p[0], p[1]};
                b[fn] = tv;
            }
            #pragma unroll
            for (int fm = 0; fm < 4; ++fm)
                #pragma unroll
                for (int fn = 0; fn < 2; ++fn)
                    acc[fm][fn] = __builtin_amdgcn_wmma_f32_16x16x4_f32(
                        false, a[fm], false, b[fn], (short)0, acc[fm][fn],
                        false, false);
        }
        __syncthreads();                       // done reading 'cur' -> refillable
    }

    // C/D layout: VGPR v, lanes 0-15 -> M=v, lanes 16-31 -> M=v+8; N = lane&15
    const float scale = 0.03125f;   // 1/sqrt(1024)
    const int mhalf = (lane >> 4) * 8;
    #pragma unroll
    for (int fm = 0; fm < 4; ++fm)
        #pragma unroll
        for (int fn = 0; fn < 2; ++fn)
            #pragma unroll
            for (int vv = 0; vv < 8; ++vv) {
                int row = bm + wm0 + fm * 16 + vv + mhalf;
                int col = bn + wn0 + fn * 16 + (lane & 15);
                S[(size_t)row * N_DIM + col] = acc[fm][fn][vv] * scale;
            }
}

// ---------------------------------------------------------------------------
// In-place per-row: K = exp(-0.1 * (1 - softmax(sim)))   (M = -10*ln K later)
// ---------------------------------------------------------------------------
__global__ __launch_bounds__(256)
void ETP_softmax_to_K(float* __restrict__ S) {
    const int row = blockIdx.x;
    const int tid = threadIdx.x;
    float* rp = S + (size_t)row * N_DIM;

    float vals[16];
    float mx = -3.4e38f;
    #pragma unroll
    for (int i = 0; i < 16; ++i) {
        vals[i] = rp[tid + i * 256];
        mx = fmaxf(mx, vals[i]);
    }
    __shared__ float red[256];
    red[tid] = mx; __syncthreads();
    for (int s = 128; s > 0; s >>= 1) {
        if (tid < s) red[tid] = fmaxf(red[tid], red[tid + s]);
        __syncthreads();
    }
    mx = red[0]; __syncthreads();

    float sum = 0.0f;
    #pragma unroll
    for (int i = 0; i < 16; ++i) {
        vals[i] = __expf(vals[i] - mx);
        sum += vals[i];
    }
    red[tid] = sum; __syncthreads();
    for (int s = 128; s > 0; s >>= 1) {
        if (tid < s) red[tid] += red[tid + s];
        __syncthreads();
    }
    const float inv = 1.0f / red[0];

    #pragma unroll
    for (int i = 0; i < 16; ++i) {
        float p = vals[i] * inv;                          // softmax
        rp[tid + i * 256] = __expf(-0.1f * (1.0f - p));   // K
    }
}

// ---------------------------------------------------------------------------
// Sinkhorn iteration kernels (K is L2-resident: 64 MB of 192 MB)
// ---------------------------------------------------------------------------
__global__ void ETP_init(float* u, float* v, float* tmp, float* loss) {
    int i = blockIdx.x * 256 + threadIdx.x;
    if (i < N_DIM) {
        u[i]   = 1.0f / (float)N_DIM;
        v[i]   = 0.0f;
        tmp[i] = 0.0f;
    }
    if (i == 0) loss[0] = 0.0f;
}

// tmp[j] += sum over 256-row chunk of K[i][j]*u[i]   (grid 16x16, coalesced)
__global__ __launch_bounds__(256)
void ETP_ktu_partial(const float* __restrict__ K, const float* __restrict__ u,
                     float* __restrict__ tmp) {
    const int j  = blockIdx.x * 256 + threadIdx.x;
    const int r0 = blockIdx.y * 256;
    float s = 0.0f;
    #pragma unroll 4
    for (int i = 0; i < 256; ++i)
        s = fmaf(K[(size_t)(r0 + i) * N_DIM + j], u[r0 + i], s);
    atomicAdd(&tmp[j], s);
}

// v = b / (K^T u + eps); rezero tmp for next iteration
__global__ __launch_bounds__(256)
void ETP_finalize_v(float* __restrict__ tmp, float* __restrict__ v) {
    int j = blockIdx.x * 256 + threadIdx.x;
    v[j] = (1.0f / (float)N_DIM) / (tmp[j] + 1e-9f);
    tmp[j] = 0.0f;
}

// u[i] = a / (K[i,:]·v + eps), wave per row (wave32 shuffle reduce)
__global__ __launch_bounds__(256)
void ETP_update_u(const float* __restrict__ K, const float* __restrict__ v,
                  float* __restrict__ u) {
    const int wave = threadIdx.x >> 5;
    const int lane = threadIdx.x & 31;
    const int row  = blockIdx.x * 8 + wave;
    const float* kr = K + (size_t)row * N_DIM;
    float s = 0.0f;
    for (int j = lane; j < N_DIM; j += 32) {
        __builtin_prefetch(kr + j + 256, 0, 1);    // global_prefetch_b8
        s = fmaf(kr[j], v[j], s);
    }
    #pragma unroll
    for (int off = 16; off > 0; off >>= 1)
        s += __shfl_down(s, off, 32);
    if (lane == 0)
        u[row] = (1.0f / (float)N_DIM) / (s + 1e-9f);
}

// ---------------------------------------------------------------------------
// transp = u * K * v^T -> out+1 ; loss = sum(transp * M), M = -10 ln K -> out[0]
// ---------------------------------------------------------------------------
__global__ __launch_bounds__(256)
void ETP_transp_loss(const float* __restrict__ K, const float* __restrict__ u,
                     const float* __restrict__ v, float* __restrict__ out) {
    const int row = blockIdx.x;
    const int tid = threadIdx.x;
    const float ui = u[row];
    const float* kr = K + (size_t)row * N_DIM;
    float* tr = out + 1 + (size_t)row * N_DIM;

    float lsum = 0.0f;
    #pragma unroll
    for (int jj = 0; jj < 16; ++jj) {
        int j = tid + jj * 256;
        float kij = kr[j];
        float t = ui * kij * v[j];
        tr[j] = t;
        lsum = fmaf(t, -10.0f * __logf(kij), lsum);
    }
    __shared__ float red[256];
    red[tid] = lsum; __syncthreads();
    for (int s = 128; s > 0; s >>= 1) {
        if (tid < s) red[tid] += red[tid + s];
        __syncthreads();
    }
    if (tid == 0) atomicAdd(out, red[0]);
}

// ---------------------------------------------------------------------------
extern "C" void kernel_launch(void* const* d_in, const int* in_sizes, int n_in,
                              void* d_out, int out_size, void* d_ws, size_t ws_size,
                              hipStream_t stream) {
    const float* X = (const float*)d_in[0];
    const float* Y = (const float*)d_in[1];
    float* out = (float*)d_out;

    float* K   = (float*)d_ws;                    // 4096*4096 (64 MB, L2-resident)
    float* u   = K + (size_t)N_DIM * N_DIM;       // 4096
    float* v   = u + N_DIM;                       // 4096
    float* tmp = v + N_DIM;                       // 4096

    dim3 gemm_grid(N_DIM / TILE_M, N_DIM / TILE_N);
    ETP_sim_gemm<<<gemm_grid, 256, 0, stream>>>(X, Y, K);      // sim into K buffer
    ETP_softmax_to_K<<<N_DIM, 256, 0, stream>>>(K);            // in-place -> K
    ETP_init<<<N_DIM / 256, 256, 0, stream>>>(u, v, tmp, out);

    for (int it = 0; it < 100; ++it) {
        ETP_ktu_partial<<<dim3(N_DIM / 256, N_DIM / 256), 256, 0, stream>>>(K, u, tmp);
        ETP_finalize_v<<<N_DIM / 256, 256, 0, stream>>>(tmp, v);
        ETP_update_u<<<N_DIM / 8, 256, 0, stream>>>(K, v, u);
    }

    ETP_transp_loss<<<N_DIM, 256, 0, stream>>>(K, u, v, out);
}